// PhaGatModel_33741263078267
// MI455X (gfx1250) — compile-verified
//
#include <hip/hip_runtime.h>
#include <math.h>

typedef __attribute__((ext_vector_type(2))) float v2f;
typedef __attribute__((ext_vector_type(8))) float v8f;

#define NEG_INF_F (-1e9f)

// const-block layout (floats) inside workspace
#define OFF_USRC 0    // 8: W_gat @ a_src
#define OFF_UDST 8    // 8: W_gat @ a_dst
#define OFF_V9   16   // 9: W_dist @ u_dst
#define OFF_C0   25   // 1: b_dist . u_dst
#define OFF_BEMB 32   // packed 2-chunk B: W_emb
#define OFF_BM1  224  // packed 3-chunk B: [W_dist@W_gat ; b_dist@W_gat ; 0; 0]
#define OFF_BGAT 416  // packed 2-chunk B: W_gat
#define CST_SIZE 608

__device__ __forceinline__ float wave_max32(float v) {
#pragma unroll
  for (int o = 16; o >= 1; o >>= 1) v = fmaxf(v, __shfl_xor(v, o, 32));
  return v;
}
__device__ __forceinline__ float wave_sum32(float v) {
#pragma unroll
  for (int o = 16; o >= 1; o >>= 1) v += __shfl_xor(v, o, 32);
  return v;
}

// ---------------------------------------------------------------------------
// Precompute (1 thread): fold tiny matrices; pack WMMA B-tiles in lane-pair
// order:  Bp[((c*2+half)*16 + ncol)*2 + j] = Blogical[c*4+half*2+j][ncol]
// so each wave-lane fetches its two B values with ONE float2 load per chunk.
// Also zero pad row 0 of both h buffers.
// ---------------------------------------------------------------------------
__global__ void precompute_kernel(const float* __restrict__ W_emb,
                                  const float* __restrict__ b_emb,
                                  const float* __restrict__ W_dist,
                                  const float* __restrict__ b_dist,
                                  const float* __restrict__ W_gat,
                                  const float* __restrict__ a_src,
                                  const float* __restrict__ a_dst,
                                  float* __restrict__ cst,
                                  float* __restrict__ hA,
                                  float* __restrict__ hB) {
  if (threadIdx.x != 0 || blockIdx.x != 0) return;
  float us[8], ud[8];
  for (int i = 0; i < 8; ++i) {
    float s0 = 0.f, s1 = 0.f;
    for (int j = 0; j < 8; ++j) {
      s0 += W_gat[i * 8 + j] * a_src[j];
      s1 += W_gat[i * 8 + j] * a_dst[j];
    }
    us[i] = s0; ud[i] = s1;
    cst[OFF_USRC + i] = s0; cst[OFF_UDST + i] = s1;
  }
  for (int i = 0; i < 9; ++i) {
    float s = 0.f;
    for (int j = 0; j < 8; ++j) s += W_dist[i * 8 + j] * ud[j];
    cst[OFF_V9 + i] = s;
  }
  {
    float s = 0.f;
    for (int j = 0; j < 8; ++j) s += b_dist[j] * ud[j];
    cst[OFF_C0] = s;
  }
  // logical M1ext rows 0..8 = W_dist@W_gat, row 9 = b_dist@W_gat
  float M1v[10][8];
  for (int r = 0; r < 9; ++r)
    for (int c = 0; c < 8; ++c) {
      float v = 0.f;
      for (int t = 0; t < 8; ++t) v += W_dist[r * 8 + t] * W_gat[t * 8 + c];
      M1v[r][c] = v;
    }
  for (int c = 0; c < 8; ++c) {
    float v = 0.f;
    for (int t = 0; t < 8; ++t) v += b_dist[t] * W_gat[t * 8 + c];
    M1v[9][c] = v;
  }
  // pack B_emb (2 chunks, logical rows 0..7 = W_emb)
  for (int c = 0; c < 2; ++c)
    for (int half = 0; half < 2; ++half)
      for (int ncol = 0; ncol < 16; ++ncol)
        for (int j = 0; j < 2; ++j) {
          int r = c * 4 + half * 2 + j;
          float v = (ncol < 8) ? W_emb[r * 8 + ncol] : 0.f;
          cst[OFF_BEMB + ((c * 2 + half) * 16 + ncol) * 2 + j] = v;
        }
  // pack B_m1 (3 chunks, logical rows 0..9 = M1v, rows 10..11 = 0)
  for (int c = 0; c < 3; ++c)
    for (int half = 0; half < 2; ++half)
      for (int ncol = 0; ncol < 16; ++ncol)
        for (int j = 0; j < 2; ++j) {
          int r = c * 4 + half * 2 + j;
          float v = (ncol < 8 && r < 10) ? M1v[r][ncol] : 0.f;
          cst[OFF_BM1 + ((c * 2 + half) * 16 + ncol) * 2 + j] = v;
        }
  // pack B_gat (2 chunks, logical rows 0..7 = W_gat)
  for (int c = 0; c < 2; ++c)
    for (int half = 0; half < 2; ++half)
      for (int ncol = 0; ncol < 16; ++ncol)
        for (int j = 0; j < 2; ++j) {
          int r = c * 4 + half * 2 + j;
          float v = (ncol < 8) ? W_gat[r * 8 + ncol] : 0.f;
          cst[OFF_BGAT + ((c * 2 + half) * 16 + ncol) * 2 + j] = v;
        }
  for (int i = 0; i < 8; ++i) { hA[i] = 0.f; hB[i] = 0.f; }
}

// ---------------------------------------------------------------------------
// Dense projection via chained V_WMMA_F32_16X16X4_F32, branch-free inner loop.
// One wave = one 16-row tile.  A: lanes 0-15 hold M=lane (K=cb,cb+1), lanes
// 16-31 hold K=cb+2,cb+3 -> one float2 load.  B pre-packed the same way.
// Epilogue adds optional per-column bias / ELU; full tiles take a uniform
// branch with 8 immediate-offset stores off a single base address.
// ---------------------------------------------------------------------------
template <int NCHUNK, bool ADD_BIAS, bool DO_ELU>
__global__ __launch_bounds__(256) void project_kernel(
    const float* __restrict__ in, int stride, const float* __restrict__ Bp,
    const float* __restrict__ bias, float* __restrict__ out, int nrows) {
  int wave = blockIdx.x * (blockDim.x >> 5) + (threadIdx.x >> 5);
  int lane = threadIdx.x & 31;
  if (wave * 16 >= nrows) return;  // uniform per wave
  int half = lane >> 4;
  int l15 = lane & 15;
  int mrow = wave * 16 + l15;
  int mrowc = (mrow < nrows) ? mrow : (nrows - 1);  // branchless clamp
  v8f acc = {0.f, 0.f, 0.f, 0.f, 0.f, 0.f, 0.f, 0.f};
#pragma unroll
  for (int c = 0; c < NCHUNK; ++c) {
    int cb = c * 4 + half * 2;
    float2 av = *(const float2*)(in + (size_t)mrowc * stride + cb);
    float2 bv = *(const float2*)(Bp + (((size_t)(c * 2 + half)) * 16 + l15) * 2);
    v2f a; a[0] = av.x; a[1] = av.y;
    v2f b; b[0] = bv.x; b[1] = bv.y;
    acc = __builtin_amdgcn_wmma_f32_16x16x4_f32(false, a, false, b, (short)0,
                                                acc, false, false);
  }
  if (l15 < 8) {
    float bb = ADD_BIAS ? bias[l15] : 0.f;
    float xv[8];
#pragma unroll
    for (int v = 0; v < 8; ++v) {
      float x = acc[v] + bb;
      if (DO_ELU) x = (x > 0.f) ? x : (__expf(x) - 1.0f);
      xv[v] = x;
    }
    int rbase = wave * 16 + half * 8;  // rows rbase..rbase+7 for this lane-half
    float* op = out + (size_t)rbase * 8 + l15;
    if (wave * 16 + 16 <= nrows) {     // full tile: uniform fast path
#pragma unroll
      for (int v = 0; v < 8; ++v) op[v * 8] = xv[v];
    } else {                            // ragged last tile only
#pragma unroll
      for (int v = 0; v < 8; ++v)
        if (rbase + v < nrows) op[v * 8] = xv[v];
    }
  }
}

// ---------------------------------------------------------------------------
// Iteration 1 attention: one wave per node, lanes = K neighbor slots.
// e_k = leaky( h0[see]·u_src + ([fdg,rij]·v9 + c0) ), softmax, accumulate
// xbar = [ sum_valid a_k*[fdg,rij] , sum_valid a_k , 0, 0 ]  (stride 12).
// ---------------------------------------------------------------------------
__global__ __launch_bounds__(256) void attn_iter1_kernel(
    const float* __restrict__ hcur, const float* __restrict__ fdg,
    const float* __restrict__ rij, const int* __restrict__ bsc,
    const int* __restrict__ see, const float* __restrict__ cst,
    float* __restrict__ xbar, int N, int K) {
  int wave = blockIdx.x * (blockDim.x >> 5) + (threadIdx.x >> 5);
  int lane = threadIdx.x & 31;
  if (wave >= N) return;
  float us[8];
#pragma unroll
  for (int i = 0; i < 8; ++i) us[i] = cst[OFF_USRC + i];
  float v9[9];
#pragma unroll
  for (int i = 0; i < 9; ++i) v9[i] = cst[OFF_V9 + i];
  float c0 = cst[OFF_C0];

  int iS = 0, iB = 0;
  if (lane < K) {
    iS = see[(size_t)wave * K + lane];
    iB = bsc[(size_t)wave * K + lane];
  }
  const float4* hp = (const float4*)(hcur + (size_t)iS * 8);
  float4 h0 = hp[0], h1 = hp[1];
  float s = h0.x * us[0] + h0.y * us[1] + h0.z * us[2] + h0.w * us[3] +
            h1.x * us[4] + h1.y * us[5] + h1.z * us[6] + h1.w * us[7];
  bool valid = (lane < K) && (iB > 0);
  float f0 = 0, f1 = 0, f2 = 0, f3 = 0, f4 = 0, f5 = 0, f6 = 0, f7 = 0, f8 = 0;
  float t = 0.f;
  if (valid) {
    size_t ei = (size_t)(iB - 1);
    const float4* fp = (const float4*)(fdg + ei * 8);
    float4 x0 = fp[0], x1 = fp[1];
    f0 = x0.x; f1 = x0.y; f2 = x0.z; f3 = x0.w;
    f4 = x1.x; f5 = x1.y; f6 = x1.z; f7 = x1.w;
    f8 = rij[ei];
    t = f0 * v9[0] + f1 * v9[1] + f2 * v9[2] + f3 * v9[3] + f4 * v9[4] +
        f5 * v9[5] + f6 * v9[6] + f7 * v9[7] + f8 * v9[8] + c0;
  }
  float e = s + t;
  e = (e >= 0.f) ? e : 0.2f * e;
  e = valid ? e : NEG_INF_F;
  float m = wave_max32(e);
  float p = (lane < K) ? __expf(e - m) : 0.f;
  float d = wave_sum32(p);
  float alpha = (d > 0.f) ? (p / d) : 0.f;
  float av = valid ? alpha : 0.f;
  float a0 = wave_sum32(av * f0);
  float a1 = wave_sum32(av * f1);
  float a2 = wave_sum32(av * f2);
  float a3 = wave_sum32(av * f3);
  float a4 = wave_sum32(av * f4);
  float a5 = wave_sum32(av * f5);
  float a6 = wave_sum32(av * f6);
  float a7 = wave_sum32(av * f7);
  float a8 = wave_sum32(av * f8);
  float a9 = wave_sum32(av);
  if (lane == 0) {
    float4* o = (float4*)(xbar + (size_t)wave * 12);
    o[0] = make_float4(a0, a1, a2, a3);
    o[1] = make_float4(a4, a5, a6, a7);
    o[2] = make_float4(a8, a9, 0.f, 0.f);  // zero-pad cols 10,11 for WMMA K=12
  }
}

// ---------------------------------------------------------------------------
// Iteration 2 attention: msg row = h1[edge_src[b_scope-1]] (double gather;
// h1 is ~6.5 MB -> L2-resident).  xbar = sum_valid a_k * msg_k (stride 8).
// ---------------------------------------------------------------------------
__global__ __launch_bounds__(256) void attn_iter2_kernel(
    const float* __restrict__ hcur, const int* __restrict__ esrc,
    const int* __restrict__ bsc, const int* __restrict__ see,
    const float* __restrict__ cst, float* __restrict__ xbar, int N, int K) {
  int wave = blockIdx.x * (blockDim.x >> 5) + (threadIdx.x >> 5);
  int lane = threadIdx.x & 31;
  if (wave >= N) return;
  float us[8], ud[8];
#pragma unroll
  for (int i = 0; i < 8; ++i) {
    us[i] = cst[OFF_USRC + i];
    ud[i] = cst[OFF_UDST + i];
  }
  int iS = 0, iB = 0;
  if (lane < K) {
    iS = see[(size_t)wave * K + lane];
    iB = bsc[(size_t)wave * K + lane];
  }
  const float4* hp = (const float4*)(hcur + (size_t)iS * 8);
  float4 h0 = hp[0], h1 = hp[1];
  float s = h0.x * us[0] + h0.y * us[1] + h0.z * us[2] + h0.w * us[3] +
            h1.x * us[4] + h1.y * us[5] + h1.z * us[6] + h1.w * us[7];
  bool valid = (lane < K) && (iB > 0);
  float m0 = 0, m1 = 0, m2 = 0, m3 = 0, m4 = 0, m5 = 0, m6 = 0, m7 = 0;
  float t = 0.f;
  if (valid) {
    int srcn = esrc[iB - 1];  // 0..N, row 0 of h is zero
    const float4* mp = (const float4*)(hcur + (size_t)srcn * 8);
    float4 x0 = mp[0], x1 = mp[1];
    m0 = x0.x; m1 = x0.y; m2 = x0.z; m3 = x0.w;
    m4 = x1.x; m5 = x1.y; m6 = x1.z; m7 = x1.w;
    t = m0 * ud[0] + m1 * ud[1] + m2 * ud[2] + m3 * ud[3] + m4 * ud[4] +
        m5 * ud[5] + m6 * ud[6] + m7 * ud[7];
  }
  float e = s + t;
  e = (e >= 0.f) ? e : 0.2f * e;
  e = valid ? e : NEG_INF_F;
  float mx = wave_max32(e);
  float p = (lane < K) ? __expf(e - mx) : 0.f;
  float d = wave_sum32(p);
  float alpha = (d > 0.f) ? (p / d) : 0.f;
  float av = valid ? alpha : 0.f;
  float a0 = wave_sum32(av * m0);
  float a1 = wave_sum32(av * m1);
  float a2 = wave_sum32(av * m2);
  float a3 = wave_sum32(av * m3);
  float a4 = wave_sum32(av * m4);
  float a5 = wave_sum32(av * m5);
  float a6 = wave_sum32(av * m6);
  float a7 = wave_sum32(av * m7);
  if (lane == 0) {
    float4* o = (float4*)(xbar + (size_t)wave * 8);
    o[0] = make_float4(a0, a1, a2, a3);
    o[1] = make_float4(a4, a5, a6, a7);
  }
}

// ---------------------------------------------------------------------------
// Readout: out[b,j] = sum_l h[l_scope[b,l], j]
// ---------------------------------------------------------------------------
__global__ __launch_bounds__(256) void readout_kernel(
    const float* __restrict__ h, const int* __restrict__ lsc,
    float* __restrict__ out, int Bg, int L) {
  int tid = blockIdx.x * blockDim.x + threadIdx.x;
  if (tid >= Bg * 8) return;
  int b = tid >> 3, j = tid & 7;
  float s = 0.f;
  for (int l = 0; l < L; ++l) {
    int idx = lsc[(size_t)b * L + l];
    s += h[(size_t)idx * 8 + j];
  }
  out[tid] = s;
}

// ---------------------------------------------------------------------------
extern "C" void kernel_launch(void* const* d_in, const int* in_sizes, int n_in,
                              void* d_out, int out_size, void* d_ws,
                              size_t ws_size, hipStream_t stream) {
  const float* tf    = (const float*)d_in[0];
  const float* fdg   = (const float*)d_in[1];
  const float* rij   = (const float*)d_in[2];
  const int*   bsc   = (const int*)d_in[3];
  const int*   see   = (const int*)d_in[4];
  const int*   lsc   = (const int*)d_in[5];
  const int*   esrc  = (const int*)d_in[6];
  const float* W_emb = (const float*)d_in[7];
  const float* b_emb = (const float*)d_in[8];
  const float* W_dist= (const float*)d_in[9];
  const float* b_dist= (const float*)d_in[10];
  const float* W_gat = (const float*)d_in[11];
  const float* a_src = (const float*)d_in[12];
  const float* a_dst = (const float*)d_in[13];
  float* out = (float*)d_out;

  const int N  = in_sizes[0] / 8;
  const int K  = in_sizes[3] / N;
  const int Bg = out_size / 8;
  const int L  = in_sizes[5] / Bg;

  float* ws = (float*)d_ws;
  const size_t hsz = (size_t)(N + 1) * 8;
  float* hA   = ws;                       // h0 / h2 (+pad row 0)
  float* hB   = ws + hsz;                 // h1 (+pad row 0)
  float* xbar = ws + 2 * hsz;             // N x 12 scratch
  float* cst  = xbar + (size_t)N * 12;    // CST_SIZE floats

  precompute_kernel<<<1, 32, 0, stream>>>(W_emb, b_emb, W_dist, b_dist, W_gat,
                                          a_src, a_dst, cst, hA, hB);

  const int tiles   = (N + 15) / 16;
  const int pblocks = (tiles + 7) / 8;    // 8 waves (tiles) per 256-thr block
  const int ablocks = (N + 7) / 8;        // 8 nodes (waves) per 256-thr block

  // h0 = tf @ W_emb + b_emb  (bias in epilogue; K=8 -> 2 WMMA chunks)
  project_kernel<2, true, false><<<pblocks, 256, 0, stream>>>(
      tf, 8, cst + OFF_BEMB, b_emb, hA + 8, N);
  // iteration 1
  attn_iter1_kernel<<<ablocks, 256, 0, stream>>>(hA, fdg, rij, bsc, see, cst,
                                                 xbar, N, K);
  project_kernel<3, false, true><<<pblocks, 256, 0, stream>>>(
      xbar, 12, cst + OFF_BM1, nullptr, hB + 8, N);
  // iteration 2
  attn_iter2_kernel<<<ablocks, 256, 0, stream>>>(hB, esrc, bsc, see, cst, xbar,
                                                 N, K);
  project_kernel<2, false, true><<<pblocks, 256, 0, stream>>>(
      xbar, 8, cst + OFF_BGAT, nullptr, hA + 8, N);
  // readout
  readout_kernel<<<(Bg * 8 + 255) / 256, 256, 0, stream>>>(hA, lsc, out, Bg, L);
}